// TestNet_68504728371298
// MI455X (gfx1250) — compile-verified
//
#include <hip/hip_runtime.h>
#include <hip/hip_bf16.h>
#include <math.h>

typedef float v2f __attribute__((ext_vector_type(2)));
typedef float v8f __attribute__((ext_vector_type(8)));
typedef unsigned int u32x4 __attribute__((ext_vector_type(4)));
typedef int i32x4 __attribute__((ext_vector_type(4)));
typedef int i32x8 __attribute__((ext_vector_type(8)));

#define BB 32
#define NN 1024
#define FF 16
#define LL 8

#if defined(__AMDGCN__) && defined(__has_builtin)
#if __has_builtin(__builtin_amdgcn_tensor_load_to_lds) && __has_builtin(__builtin_amdgcn_s_wait_tensorcnt)
#define USE_TDM 1
#endif
#endif

#ifdef USE_TDM
// 1D contiguous global->LDS copy via the Tensor Data Mover.
// D# group0/group1 bitfields per CDNA5 ISA ch.10.8 (data_size=4B, 1D tile).
__device__ __forceinline__ void tdm_copy_f32(const float* gsrc, unsigned lds_byte_off,
                                             unsigned nelem) {
    unsigned long long ga = (unsigned long long)(const void*)gsrc;
    u32x4 g0;
    g0[0] = 1u;                                            // count=1, user desc
    g0[1] = lds_byte_off;                                  // lds_addr
    g0[2] = (unsigned)ga;                                  // global_addr[31:0]
    g0[3] = (unsigned)((ga >> 32) & 0x1FFFFFFu) | (2u << 30); // addr[56:32] | type=2
    i32x8 g1;
    g1[0] = (int)(2u << 16);                               // data_size=2 -> 4 bytes
    g1[1] = (int)((nelem & 0xFFFFu) << 16);                // tensor_dim0[15:0]
    g1[2] = (int)((nelem >> 16) | (1u << 16));             // tensor_dim0[31:16], tensor_dim1=1
    g1[3] = (int)((nelem & 0xFFFFu) << 16);                // tile_dim0
    g1[4] = 1;                                             // tile_dim1 = 1
    g1[5] = (int)nelem;                                    // tensor_dim0_stride[31:0]
    g1[6] = 0;
    g1[7] = 0;
    i32x4 z4 = {0, 0, 0, 0};
#if __clang_major__ >= 23
    i32x8 z8 = {0, 0, 0, 0, 0, 0, 0, 0};
    __builtin_amdgcn_tensor_load_to_lds(g0, g1, z4, z4, z8, 0);
#else
    __builtin_amdgcn_tensor_load_to_lds(g0, g1, z4, z4, 0);
#endif
}
#endif

// ---------------------------------------------------------------------------
// Kernel 0: row-L2-normalize the embedding table (C=4 x F=16 = 64 values)
// ---------------------------------------------------------------------------
__global__ __launch_bounds__(64) void emb_norm_kernel(const float* __restrict__ embs,
                                                      float* __restrict__ emb_n) {
    int t = threadIdx.x;
    int c = t >> 4;
    float ss = 0.f;
#pragma unroll
    for (int f = 0; f < 16; ++f) {
        float v = embs[c * 16 + f];
        ss += v * v;
    }
    float nrm = fmaxf(sqrtf(ss), 1e-12f);
    emb_n[t] = embs[t] / nrm;
}

// ---------------------------------------------------------------------------
// Kernel 1: x0 = nodes @ emb_n   ([B*N,4] @ [4,16]) -- one WMMA per 16-row tile
// ---------------------------------------------------------------------------
__global__ __launch_bounds__(256) void x0_wmma_kernel(const float* __restrict__ nodes,
                                                      const float* __restrict__ emb_n,
                                                      float* __restrict__ x0) {
    const int tid   = threadIdx.x;
    const int wave  = tid >> 5;
    const int lane  = tid & 31;
    const int lhalf = lane >> 4;
    const int l16   = lane & 15;
    const int tile0 = (blockIdx.x * 8 + wave) * 16;
    const int row   = tile0 + l16;

    v2f a = *(const v2f*)(nodes + (size_t)row * 4 + 2 * lhalf);
    v2f b;
    b.x = emb_n[(2 * lhalf) * 16 + l16];
    b.y = emb_n[(2 * lhalf + 1) * 16 + l16];

    v8f acc = {};
    acc = __builtin_amdgcn_wmma_f32_16x16x4_f32(false, a, false, b, (short)0, acc,
                                                false, false);
#pragma unroll
    for (int r = 0; r < 8; ++r) {
        int m = tile0 + r + 8 * lhalf;
        x0[(size_t)m * FF + l16] = acc[r];
    }
}

// ---------------------------------------------------------------------------
// Kernel 2: Xw = X @ W   ([B*N,16] @ [16,16]) -- 4 WMMAs (K=16) per 16-row tile
// ---------------------------------------------------------------------------
__global__ __launch_bounds__(256) void xw_wmma_kernel(const float* __restrict__ x,
                                                      const float* __restrict__ w,
                                                      float* __restrict__ xw) {
    const int tid   = threadIdx.x;
    const int wave  = tid >> 5;
    const int lane  = tid & 31;
    const int lhalf = lane >> 4;
    const int l16   = lane & 15;
    const int tile0 = (blockIdx.x * 8 + wave) * 16;
    const int row   = tile0 + l16;

    const float* arow = x + (size_t)row * FF + 2 * lhalf;
    const float* bcol = w + (2 * lhalf) * FF + l16;

    v8f acc = {};
#pragma unroll
    for (int k0 = 0; k0 < 16; k0 += 4) {
        v2f a = *(const v2f*)(arow + k0);
        v2f b;
        b.x = bcol[k0 * FF];
        b.y = bcol[k0 * FF + FF];
        acc = __builtin_amdgcn_wmma_f32_16x16x4_f32(false, a, false, b, (short)0, acc,
                                                    false, false);
    }
#pragma unroll
    for (int r = 0; r < 8; ++r) {
        int m = tile0 + r + 8 * lhalf;
        xw[(size_t)m * FF + l16] = acc[r];
    }
}

// ---------------------------------------------------------------------------
// Kernel 3: Y = leaky_relu(Adj[b] @ Xw[b] + bias)
// Block = 256 thr (8 waves); grid = (4, B); each wave owns TWO 16x16 tiles
// (rows m0 and m0+128) sharing one B fragment: per K-step this is
// 1 ds_load_b64 + 2 global_load_b64 + 2 v_wmma_f32_16x16x4_f32.
// Xw[b] staged in LDS pair-interleaved: lds[(k>>1)*32 + n*2 + (k&1)].
// ---------------------------------------------------------------------------
__global__ __launch_bounds__(256) void gcn_layer_kernel(const float* __restrict__ adj,
                                                        const float* __restrict__ xw,
                                                        const float* __restrict__ bias,
                                                        float* __restrict__ out) {
    __shared__ float ldsX[NN * FF];   // 64 KB
    const int b   = blockIdx.y;
    const int tid = threadIdx.x;

    {   // pair-interleaved staging: 512 pairs x 4 col-groups, 8 groups/thread
        const float* src = xw + (size_t)b * NN * FF;
#pragma unroll
        for (int i = 0; i < 8; ++i) {
            int    g  = tid + i * 256;        // 0..2047
            int    p  = g >> 2;               // K-pair 0..511
            int    n0 = (g & 3) * 4;          // column group
            float4 r0 = *(const float4*)(src + p * 32 + n0);
            float4 r1 = *(const float4*)(src + p * 32 + 16 + n0);
            v2f*   d  = (v2f*)ldsX + p * 16 + n0;
            d[0] = (v2f){r0.x, r1.x};
            d[1] = (v2f){r0.y, r1.y};
            d[2] = (v2f){r0.z, r1.z};
            d[3] = (v2f){r0.w, r1.w};
        }
    }
    __syncthreads();

    const int wave  = tid >> 5;
    const int lane  = tid & 31;
    const int lhalf = lane >> 4;
    const int l16   = lane & 15;
    const int m0    = blockIdx.x * 256 + wave * 16;   // tile0 rows
    const int row0  = m0 + l16;
    const int row1  = row0 + 128;                     // tile1 rows

    const float* abase = adj + (size_t)b * NN * NN + 2 * lhalf;
    const float* arow0 = abase + (size_t)row0 * NN;
    const float* arow1 = abase + (size_t)row1 * NN;
    // B fragment: pair (k0/2 + lhalf), column l16  -> one b64 LDS load
    const v2f* bbase = (const v2f*)ldsX + lhalf * 16 + l16;

    v8f acc0 = {};
    v8f acc1 = {};
    for (int kk = 0; kk < NN; kk += 32) {
        __builtin_prefetch(arow0 + kk + 128, 0, 1);
        __builtin_prefetch(arow1 + kk + 128, 0, 1);
#pragma unroll
        for (int k0 = kk; k0 < kk + 32; k0 += 4) {
            v2f bf = bbase[(k0 >> 1) * 16];           // ds_load_b64
            v2f a0 = *(const v2f*)(arow0 + k0);       // global_load_b64
            v2f a1 = *(const v2f*)(arow1 + k0);       // global_load_b64
            acc0 = __builtin_amdgcn_wmma_f32_16x16x4_f32(false, a0, false, bf,
                                                         (short)0, acc0, false, false);
            acc1 = __builtin_amdgcn_wmma_f32_16x16x4_f32(false, a1, false, bf,
                                                         (short)0, acc1, false, false);
        }
    }

    const float bv   = bias[l16];
    float*      orow = out + (size_t)b * NN * FF;
#pragma unroll
    for (int r = 0; r < 8; ++r) {
        int   m = m0 + r + 8 * lhalf;
        float v = acc0[r] + bv;
        v       = (v > 0.f) ? v : 0.01f * v;
        orow[(size_t)m * FF + l16] = v;
    }
#pragma unroll
    for (int r = 0; r < 8; ++r) {
        int   m = m0 + 128 + r + 8 * lhalf;
        float v = acc1[r] + bv;
        v       = (v > 0.f) ? v : 0.01f * v;
        orow[(size_t)m * FF + l16] = v;
    }
}

// ---------------------------------------------------------------------------
// Kernel 4: GGNN readout. One block per batch; weights staged via TDM
// (TENSOR_LOAD_TO_LDS + s_wait_tensorcnt) when available; deterministic
// LDS tree reduction (no FP atomics -> bitwise-stable across replays).
// ---------------------------------------------------------------------------
__global__ __launch_bounds__(256) void readout_kernel(
        const float* __restrict__ x,  const float* __restrict__ x0,
        const float* __restrict__ iw1, const float* __restrict__ ib1,
        const float* __restrict__ iw2, const float* __restrict__ ib2,
        const float* __restrict__ iw3, const float* __restrict__ ib3,
        const float* __restrict__ jw1, const float* __restrict__ jb1,
        const float* __restrict__ jw3, const float* __restrict__ jb3,
        const float* __restrict__ clw, float* __restrict__ out) {
    __shared__ float Wall[5 * 256];   // W1,W2,W3,JW1,JW3
    __shared__ float Bv[5 * 16], CW[16];
    __shared__ float red[16 * 256];   // 16 KB

    const int tid = threadIdx.x;
    const int b   = blockIdx.x;

#ifdef USE_TDM
    if (tid < 32) {   // one wave issues 5 TDM descriptor loads
        unsigned base = (unsigned)(unsigned long long)(const void*)&Wall[0];
        tdm_copy_f32(iw1, base + 0 * 1024, 256);
        tdm_copy_f32(iw2, base + 1 * 1024, 256);
        tdm_copy_f32(iw3, base + 2 * 1024, 256);
        tdm_copy_f32(jw1, base + 3 * 1024, 256);
        tdm_copy_f32(jw3, base + 4 * 1024, 256);
        __builtin_amdgcn_s_wait_tensorcnt(0);
    }
#else
    Wall[0 * 256 + tid] = iw1[tid];
    Wall[1 * 256 + tid] = iw2[tid];
    Wall[2 * 256 + tid] = iw3[tid];
    Wall[3 * 256 + tid] = jw1[tid];
    Wall[4 * 256 + tid] = jw3[tid];
#endif
    if (tid < 16) {
        Bv[0 * 16 + tid] = ib1[tid];
        Bv[1 * 16 + tid] = ib2[tid];
        Bv[2 * 16 + tid] = ib3[tid];
        Bv[3 * 16 + tid] = jb1[tid];
        Bv[4 * 16 + tid] = jb3[tid];
        CW[tid]          = clw[tid];
    }
    __syncthreads();

    float acc[16];
#pragma unroll
    for (int f = 0; f < 16; ++f) acc[f] = 0.f;

    for (int n = tid; n < NN; n += 256) {
        const float* xr  = x  + ((size_t)b * NN + n) * FF;
        const float* x0r = x0 + ((size_t)b * NN + n) * FF;
        float xv[16], x0v[16], av[16], jav[16];
#pragma unroll
        for (int f = 0; f < 16; ++f) { xv[f] = xr[f]; x0v[f] = x0r[f]; }

#pragma unroll
        for (int j = 0; j < 16; ++j) {
            float s1 = Bv[0 * 16 + j], s2 = Bv[1 * 16 + j], s4 = Bv[3 * 16 + j];
#pragma unroll
            for (int f = 0; f < 16; ++f) {
                s1 += xv[f]  * Wall[0 * 256 + f * 16 + j];
                s2 += x0v[f] * Wall[1 * 256 + f * 16 + j];
                s4 += xv[f]  * Wall[3 * 256 + f * 16 + j];
            }
            av[j]  = fmaxf(s1, 0.f) + fmaxf(s2, 0.f);
            jav[j] = fmaxf(s4, 0.f);
        }
#pragma unroll
        for (int j = 0; j < 16; ++j) {
            float s3 = Bv[2 * 16 + j], s5 = Bv[4 * 16 + j];
#pragma unroll
            for (int f = 0; f < 16; ++f) {
                s3 += av[f]  * Wall[2 * 256 + f * 16 + j];
                s5 += jav[f] * Wall[4 * 256 + f * 16 + j];
            }
            float iv = 1.f / (1.f + expf(-s3));
            float jv = tanhf(s5);
            acc[j] += iv * jv;
        }
    }

#pragma unroll
    for (int f = 0; f < 16; ++f) red[f * 256 + tid] = acc[f];
    __syncthreads();
    for (int s = 128; s > 0; s >>= 1) {
        if (tid < s) {
#pragma unroll
            for (int f = 0; f < 16; ++f) red[f * 256 + tid] += red[f * 256 + tid + s];
        }
        __syncthreads();
    }
    if (tid == 0) {
        float o = 0.f;
#pragma unroll
        for (int f = 0; f < 16; ++f) o += red[f * 256] * CW[f];
        out[b] = o;
    }
}

// ---------------------------------------------------------------------------
extern "C" void kernel_launch(void* const* d_in, const int* in_sizes, int n_in,
                              void* d_out, int out_size, void* d_ws, size_t ws_size,
                              hipStream_t stream) {
    const float* nodes  = (const float*)d_in[0];
    const float* adj    = (const float*)d_in[1];
    const float* embs   = (const float*)d_in[2];
    const float* conv_w = (const float*)d_in[3];
    const float* conv_b = (const float*)d_in[4];
    const float* i_w1   = (const float*)d_in[5];
    const float* i_b1   = (const float*)d_in[6];
    const float* i_w2   = (const float*)d_in[7];
    const float* i_b2   = (const float*)d_in[8];
    const float* i_w3   = (const float*)d_in[9];
    const float* i_b3   = (const float*)d_in[10];
    const float* j_w1   = (const float*)d_in[11];
    const float* j_b1   = (const float*)d_in[12];
    const float* j_w3   = (const float*)d_in[13];
    const float* j_b3   = (const float*)d_in[14];
    const float* cl_w   = (const float*)d_in[15];

    const size_t XSZ = (size_t)BB * NN * FF;
    float* ws    = (float*)d_ws;
    float* emb_n = ws;
    float* x0    = ws + 256;
    float* xA    = x0 + XSZ;
    float* xB    = xA + XSZ;
    float* xw    = xB + XSZ;

    emb_norm_kernel<<<1, 64, 0, stream>>>(embs, emb_n);
    x0_wmma_kernel<<<dim3(256), 256, 0, stream>>>(nodes, emb_n, x0);

    const float* cur = x0;
    float*       nxt = xA;
    for (int l = 0; l < LL; ++l) {
        xw_wmma_kernel<<<dim3(256), 256, 0, stream>>>(cur, conv_w + l * 256, xw);
        gcn_layer_kernel<<<dim3(4, BB), 256, 0, stream>>>(adj, xw, conv_b + l * 16, nxt);
        cur = nxt;
        nxt = (nxt == xA) ? xB : xA;
    }

    readout_kernel<<<BB, 256, 0, stream>>>(cur, x0,
                                           i_w1, i_b1, i_w2, i_b2, i_w3, i_b3,
                                           j_w1, j_b1, j_w3, j_b3, cl_w,
                                           (float*)d_out);
}